// MNIST_GCN_65721589563632
// MI455X (gfx1250) — compile-verified
//
#include <hip/hip_runtime.h>
#include <math.h>

typedef __attribute__((ext_vector_type(2))) float v2f;
typedef __attribute__((ext_vector_type(8))) float v8f;

#define N_GRAPHS 1024
#define POOL_F   32

// ---------------------------------------------------------------------------
// Degree / normalization
// ---------------------------------------------------------------------------
__global__ void k_fill1(float* __restrict__ p, int n) {
    int i = blockIdx.x * blockDim.x + threadIdx.x;
    if (i < n) p[i] = 1.0f;  // self-loop pre-counted
}

__global__ void k_deg_edges(const int* __restrict__ dst, float* __restrict__ deg, long long nE) {
    long long i = (long long)blockIdx.x * blockDim.x + threadIdx.x;
    if (i < nE) unsafeAtomicAdd(&deg[dst[i]], 1.0f);
}

__global__ void k_rsqrt_inplace(float* __restrict__ p, int n) {
    int i = blockIdx.x * blockDim.x + threadIdx.x;
    if (i < n) p[i] = rsqrtf(p[i]);  // deg >= 1 always (self-loop)
}

// ---------------------------------------------------------------------------
// FP32 WMMA GEMM:  H[nRows x Nout] = X[nRows x K] @ W[K x Nout]
// One wave per 16x16 output tile, K-loop in steps of 4 using
// V_WMMA_F32_16X16X4_F32.  Fragment layouts per CDNA5 ISA 7.12.2:
//   A 16x4 : lanes 0-15 -> {K=k, k+1}, lanes 16-31 -> {K=k+2, k+3}, M = lane%16
//   B 4x16 : same K split across lane halves, N = lane%16
//   C/D    : VGPR r -> rows (tileM*16 + half*8 + r), col = lane%16
// Requires nRows % 16 == 0 and Nout % 16 == 0 (true here: 200000, 64/32).
// ---------------------------------------------------------------------------
__global__ __launch_bounds__(256) void k_gemm_wmma_f32(
    const float* __restrict__ X, const float* __restrict__ W,
    float* __restrict__ H, int nRows, int K, int Nout)
{
    const int lane  = threadIdx.x & 31;
    const int wib   = threadIdx.x >> 5;
    const int tilesN = Nout >> 4;
    const long long task  = (long long)blockIdx.x * 8 + wib;
    const long long total = (long long)(nRows >> 4) * tilesN;
    if (task >= total) return;                 // whole-wave uniform exit
    const int tileM = (int)(task / tilesN);
    const int tileN = (int)(task % tilesN);
    const int half  = lane >> 4;               // 0: K=k,k+1   1: K=k+2,k+3
    const int lr    = lane & 15;
    const int arow  = tileM * 16 + lr;
    const int bcol  = tileN * 16 + lr;

    const float* __restrict__ xrow = X + (long long)arow * K;
    v8f c = {0.f, 0.f, 0.f, 0.f, 0.f, 0.f, 0.f, 0.f};

    for (int k = 0; k < K; k += 4) {
        v2f a, b;
        a.x = xrow[k + 2 * half + 0];
        a.y = xrow[k + 2 * half + 1];
        b.x = W[(long long)(k + 2 * half + 0) * Nout + bcol];
        b.y = W[(long long)(k + 2 * half + 1) * Nout + bcol];
        c = __builtin_amdgcn_wmma_f32_16x16x4_f32(
                /*neg_a=*/false, a, /*neg_b=*/false, b,
                /*c_mod=*/(short)0, c, /*reuse_a=*/false, /*reuse_b=*/false);
    }

    float* __restrict__ hp = H + (long long)(tileM * 16 + half * 8) * Nout + bcol;
#pragma unroll
    for (int r = 0; r < 8; ++r) hp[(long long)r * Nout] = c[r];
}

// ---------------------------------------------------------------------------
// Aggregation: self-loop init, edge scatter-add, bias+activation
// ---------------------------------------------------------------------------
template <int F>
__global__ void k_agg_init(const float* __restrict__ H, const float* __restrict__ dinv,
                           float* __restrict__ A, int nNodes)
{
    long long gid = (long long)blockIdx.x * blockDim.x + threadIdx.x;
    long long total = (long long)nNodes * F;
    if (gid >= total) return;
    int n = (int)(gid / F);
    float di = dinv[n];
    A[gid] = H[gid] * di * di;                 // self-loop message
}

template <int F>  // F = 64 or 32; consecutive lanes handle adjacent float4 chunks of one edge
__global__ void k_agg_edges(const int* __restrict__ src, const int* __restrict__ dst,
                            const float* __restrict__ dinv, const float* __restrict__ H,
                            float* __restrict__ A, long long nE)
{
    constexpr int FC = F / 4;
    long long gid = (long long)blockIdx.x * blockDim.x + threadIdx.x;
    if (gid >= nE * FC) return;
    long long e = gid / FC;                    // FC is power of two -> shift
    int cidx = (int)(gid % FC);
    int s = src[e], d = dst[e];
    float w = dinv[s] * dinv[d];
    float4 v = ((const float4*)(H + (long long)s * F))[cidx];
    float* ap = A + (long long)d * F + cidx * 4;
    unsafeAtomicAdd(ap + 0, v.x * w);
    unsafeAtomicAdd(ap + 1, v.y * w);
    unsafeAtomicAdd(ap + 2, v.z * w);
    unsafeAtomicAdd(ap + 3, v.w * w);
}

__global__ void k_bias_act(float* __restrict__ A, const float* __restrict__ b,
                           int F, int actTanh, long long total)
{
    long long gid = (long long)blockIdx.x * blockDim.x + threadIdx.x;
    if (gid >= total) return;
    int f = (int)(gid % F);
    float v = A[gid] + b[f];
    A[gid] = actTanh ? tanhf(v) : fmaxf(v, 0.0f);
}

// ---------------------------------------------------------------------------
// Pooling + head
// ---------------------------------------------------------------------------
__global__ void k_zero(float* __restrict__ p, int n) {
    int i = blockIdx.x * blockDim.x + threadIdx.x;
    if (i < n) p[i] = 0.0f;
}

__global__ void k_pool(const int* __restrict__ batch, const float* __restrict__ Hl,
                       float* __restrict__ gmax, float* __restrict__ gsum,
                       float* __restrict__ cnt, int nNodes)
{
    int gid = blockIdx.x * blockDim.x + threadIdx.x;   // nNodes * 8 chunks
    int n = gid >> 3, cidx = gid & 7;
    if (n >= nNodes) return;
    int g = batch[n];
    float4 v = ((const float4*)(Hl + (long long)n * POOL_F))[cidx];
    // post-ReLU values are >= 0 -> uint compare == float compare
    unsigned int* gm = (unsigned int*)(gmax + g * POOL_F + cidx * 4);
    atomicMax(gm + 0, __float_as_uint(v.x));
    atomicMax(gm + 1, __float_as_uint(v.y));
    atomicMax(gm + 2, __float_as_uint(v.z));
    atomicMax(gm + 3, __float_as_uint(v.w));
    float* gs = gsum + g * POOL_F + cidx * 4;
    unsafeAtomicAdd(gs + 0, v.x);
    unsafeAtomicAdd(gs + 1, v.y);
    unsafeAtomicAdd(gs + 2, v.z);
    unsafeAtomicAdd(gs + 3, v.w);
    if (cidx == 0) unsafeAtomicAdd(&cnt[g], 1.0f);
}

__global__ void k_head(const float* __restrict__ gmax, const float* __restrict__ gsum,
                       const float* __restrict__ cnt, const float* __restrict__ Wout,
                       const float* __restrict__ bout, float* __restrict__ out)
{
    int gid = blockIdx.x * blockDim.x + threadIdx.x;   // N_GRAPHS * 10
    if (gid >= N_GRAPHS * 10) return;
    int g = gid / 10, o = gid % 10;
    float inv = 1.0f / fmaxf(cnt[g], 1.0f);
    float acc = bout[o];
#pragma unroll 4
    for (int f = 0; f < POOL_F; ++f)
        acc += gmax[g * POOL_F + f] * Wout[f * 10 + o];
#pragma unroll 4
    for (int f = 0; f < POOL_F; ++f)
        acc += gsum[g * POOL_F + f] * inv * Wout[(POOL_F + f) * 10 + o];
    out[gid] = acc;
}

// ---------------------------------------------------------------------------
// Host driver
// ---------------------------------------------------------------------------
static inline int blks(long long total, int t) { return (int)((total + t - 1) / t); }

extern "C" void kernel_launch(void* const* d_in, const int* in_sizes, int n_in,
                              void* d_out, int out_size, void* d_ws, size_t ws_size,
                              hipStream_t stream)
{
    const float* x     = (const float*)d_in[0];
    const int*   eidx  = (const int*)  d_in[1];
    const int*   batch = (const int*)  d_in[2];
    const float* W0    = (const float*)d_in[3];
    const float* b0    = (const float*)d_in[4];
    const float* W1    = (const float*)d_in[5];
    const float* b1    = (const float*)d_in[6];
    const float* W2    = (const float*)d_in[7];
    const float* b2    = (const float*)d_in[8];
    const float* W3    = (const float*)d_in[9];
    const float* b3    = (const float*)d_in[10];
    const float* Wout  = (const float*)d_in[11];
    const float* bout  = (const float*)d_in[12];

    const int       N = in_sizes[0] / 8;       // NUM_FEATURES = 8
    const long long E = (long long)in_sizes[1] / 2;
    const int* src = eidx;
    const int* dst = eidx + E;

    float* ws   = (float*)d_ws;
    float* dinv = ws;                                  // N
    float* H    = dinv + N;                            // N * 64
    float* A    = H    + (size_t)N * 64;               // N * 64
    float* gmax = A    + (size_t)N * 64;               // 1024 * 32
    float* gsum = gmax + N_GRAPHS * POOL_F;            // 1024 * 32
    float* cnt  = gsum + N_GRAPHS * POOL_F;            // 1024

    const int T = 256;

    // --- symmetric normalization dinv = rsqrt(in_degree + 1) ---
    k_fill1<<<blks(N, T), T, 0, stream>>>(dinv, N);
    k_deg_edges<<<blks(E, T), T, 0, stream>>>(dst, dinv, E);
    k_rsqrt_inplace<<<blks(N, T), T, 0, stream>>>(dinv, N);

    // --- layer 0: 8 -> 64, tanh ---
    {
        long long tiles = (long long)(N >> 4) * (64 >> 4);
        k_gemm_wmma_f32<<<blks(tiles, 8), T, 0, stream>>>(x, W0, H, N, 8, 64);
        k_agg_init<64><<<blks((long long)N * 64, T), T, 0, stream>>>(H, dinv, A, N);
        k_agg_edges<64><<<blks(E * 16, T), T, 0, stream>>>(src, dst, dinv, H, A, E);
        k_bias_act<<<blks((long long)N * 64, T), T, 0, stream>>>(A, b0, 64, 1, (long long)N * 64);
    }
    // --- layer 1: 64 -> 64, relu ---
    {
        long long tiles = (long long)(N >> 4) * (64 >> 4);
        k_gemm_wmma_f32<<<blks(tiles, 8), T, 0, stream>>>(A, W1, H, N, 64, 64);
        k_agg_init<64><<<blks((long long)N * 64, T), T, 0, stream>>>(H, dinv, A, N);
        k_agg_edges<64><<<blks(E * 16, T), T, 0, stream>>>(src, dst, dinv, H, A, E);
        k_bias_act<<<blks((long long)N * 64, T), T, 0, stream>>>(A, b1, 64, 0, (long long)N * 64);
    }
    // --- layer 2: 64 -> 32, relu ---
    {
        long long tiles = (long long)(N >> 4) * (32 >> 4);
        k_gemm_wmma_f32<<<blks(tiles, 8), T, 0, stream>>>(A, W2, H, N, 64, 32);
        k_agg_init<32><<<blks((long long)N * 32, T), T, 0, stream>>>(H, dinv, A, N);
        k_agg_edges<32><<<blks(E * 8, T), T, 0, stream>>>(src, dst, dinv, H, A, E);
        k_bias_act<<<blks((long long)N * 32, T), T, 0, stream>>>(A, b2, 32, 0, (long long)N * 32);
    }
    // --- layer 3: 32 -> 32, relu ---
    {
        long long tiles = (long long)(N >> 4) * (32 >> 4);
        k_gemm_wmma_f32<<<blks(tiles, 8), T, 0, stream>>>(A, W3, H, N, 32, 32);
        k_agg_init<32><<<blks((long long)N * 32, T), T, 0, stream>>>(H, dinv, A, N);
        k_agg_edges<32><<<blks(E * 8, T), T, 0, stream>>>(src, dst, dinv, H, A, E);
        k_bias_act<<<blks((long long)N * 32, T), T, 0, stream>>>(A, b3, 32, 0, (long long)N * 32);
    }

    // --- pooling + head ---
    k_zero<<<blks(N_GRAPHS * POOL_F * 2 + N_GRAPHS, T), T, 0, stream>>>(gmax, N_GRAPHS * POOL_F * 2 + N_GRAPHS);
    k_pool<<<blks((long long)N * 8, T), T, 0, stream>>>(batch, A, gmax, gsum, cnt, N);
    k_head<<<blks(N_GRAPHS * 10, T), T, 0, stream>>>(gmax, gsum, cnt, Wout, bout, (float*)d_out);
}